// Attention_51007031607883
// MI455X (gfx1250) — compile-verified
//
#include <hip/hip_runtime.h>
#include <math.h>

// ---------------------------------------------------------------------------
// Additive (Bahdanau) attention for MI455X / gfx1250, wave32.
//   B=4, Q=256, C=1024, QD=CD=512, H=128
// Pipeline:
//   1) mc = context @ wc_w^T + wc_b          (WMMA f32 GEMM, 4096x128, K=512)
//   2) mq = query   @ wq_w^T + wq_b          (WMMA f32 GEMM, 1024x128, K=512)
//   3) emission/mask/softmax fused per (b,q) (v_tanh_f32 / v_exp_f32) -> attn
//   4) wc = attn @ context (batched)          (WMMA f32 GEMM, 256x512, K=1024)
//   5) out = tanh([wc|query] @ lo_w^T + lo_b) (WMMA f32 GEMM, split-A, K=1024)
// ---------------------------------------------------------------------------

typedef __attribute__((ext_vector_type(2))) float v2f;
typedef __attribute__((ext_vector_type(8))) float v8f;

#define B_  4
#define Q_  256
#define C_  1024
#define QD_ 512
#define CD_ 512
#define H_  128

// Hardware tanh (gfx1250 VOP1 transcendental v_tanh_f32).
// Prefer the clang builtin (compiler handles the TRANS RAW hazard by
// scheduling an independent op into the slot, so tanh co-executes with the
// surrounding FMAs). Fall back to inline asm with an explicit v_nop to cover
// the hazard the compiler cannot see through an asm block.
#if defined(__has_builtin)
#if __has_builtin(__builtin_amdgcn_tanhf)
#define HAVE_TANH_BUILTIN 1
__device__ __forceinline__ float fast_tanh(float x) {
  return __builtin_amdgcn_tanhf(x);
}
#elif __has_builtin(__builtin_amdgcn_tanh_f32)
#define HAVE_TANH_BUILTIN 1
__device__ __forceinline__ float fast_tanh(float x) {
  return __builtin_amdgcn_tanh_f32(x);
}
#endif
#endif
#ifndef HAVE_TANH_BUILTIN
__device__ __forceinline__ float fast_tanh(float x) {
  float y;
  asm volatile("v_tanh_f32 %0, %1\n\tv_nop" : "=v"(y) : "v"(x));
  return y;
}
#endif

// ---------------------------------------------------------------------------
// Generic fp32 WMMA GEMM:  Y = A @ B (+bias) (optional tanh)
//   A: (M,K) row-major. If SPLIT_A, logical A columns [0,K/2) come from A0
//      (row-major, ld=K/2) and [K/2,K) from A1 (row-major, ld=K/2).
//   B: if B_IS_NK, Bm is (N,K) row-major (nn.Linear weight, y = x W^T);
//      else Bm is (K,N) row-major.
//   One wave computes one 16x16 output tile via v_wmma_f32_16x16x4_f32.
// Fragment mapping (ISA 7.12.2, wave32):
//   A frag  a[r]: lane l -> A[tm + l%16][k0 + r + 2*(l/16)]  (float2 load)
//   B frag  b[r]: lane l -> B[k0 + r + 2*(l/16)][tn + l%16]
//   C/D     d[r]: lane l -> Y[tm + r + 8*(l/16)][tn + l%16]
// ---------------------------------------------------------------------------
template <bool B_IS_NK, bool SPLIT_A, bool HAS_BIAS, bool DO_TANH>
__global__ void wmma_gemm_f32_kernel(const float* __restrict__ A0,
                                     const float* __restrict__ A1,
                                     const float* __restrict__ Bm,
                                     const float* __restrict__ bias,
                                     float* __restrict__ Y,
                                     int M, int N, int K,
                                     int tilesN, int numTiles,
                                     long aStride, long bStride, long yStride) {
  const int wave = threadIdx.x >> 5;
  const int tile = blockIdx.x * (blockDim.x >> 5) + wave;
  if (tile >= numTiles) return;  // uniform per-wave; EXEC stays all-1 for WMMA

  const int batch = blockIdx.y;
  const float* A0b = A0 + (long)batch * aStride;
  const float* A1b = SPLIT_A ? (A1 + (long)batch * aStride) : nullptr;
  const float* Bb  = Bm + (long)batch * bStride;
  float* Yb        = Y + (long)batch * yStride;

  const int tm = (tile / tilesN) << 4;
  const int tn = (tile % tilesN) << 4;
  const int lane  = threadIdx.x & 31;
  const int l16   = lane & 15;
  const int khalf = lane >> 4;
  const int Khalf = K >> 1;

  v8f acc = {};

  const float* arow = nullptr;
  const float* arow0 = nullptr;
  const float* arow1 = nullptr;
  if (SPLIT_A) {
    arow0 = A0b + (long)(tm + l16) * Khalf;
    arow1 = A1b + (long)(tm + l16) * Khalf;
  } else {
    arow = A0b + (long)(tm + l16) * K;
  }

#pragma unroll 4
  for (int k0 = 0; k0 < K; k0 += 4) {
    const int ka = k0 + 2 * khalf;  // ka, ka+1 never straddle the K/2 split

    v2f a;
    if (SPLIT_A) {
      const float* src = (ka < Khalf) ? (arow0 + ka) : (arow1 + (ka - Khalf));
      a = *(const v2f*)src;
    } else {
      a = *(const v2f*)(arow + ka);
    }

    v2f b;
    if (B_IS_NK) {
      b = *(const v2f*)(Bb + (long)(tn + l16) * K + ka);
    } else {
      b.x = Bb[(long)ka * N + tn + l16];
      b.y = Bb[(long)(ka + 1) * N + tn + l16];
    }

    acc = __builtin_amdgcn_wmma_f32_16x16x4_f32(
        /*neg_a=*/false, a, /*neg_b=*/false, b,
        /*c_mod=*/(short)0, acc, /*reuse_a=*/false, /*reuse_b=*/false);
  }

  const float bv = HAS_BIAS ? bias[tn + l16] : 0.0f;
#pragma unroll
  for (int r = 0; r < 8; ++r) {
    const int row = tm + r + 8 * khalf;
    float v = acc[r] + bv;
    if (DO_TANH) v = fast_tanh(v);
    Yb[(long)row * N + tn + l16] = v;
  }
}

// ---------------------------------------------------------------------------
// Fused emission + mask + softmax, one block per (b,q), 256 threads.
//   e[c] = we_b + sum_h we_w[h] * tanh(mc[b,c,h] + mq[b,q,h])   (v_tanh_f32)
//   attn[b,q,:] = softmax_c(mask ? e : -inf), fully-masked rows -> 0
// Each thread owns 4 values of c (c = tid + t*256); mc rows are read as
// float4 from L2 (mc[b] is 512 KB, reused by all 256 q-blocks of batch b).
// ---------------------------------------------------------------------------
__global__ void emission_softmax_kernel(const float* __restrict__ mc,   // (B*C,H)
                                        const float* __restrict__ mq,   // (B*Q,H)
                                        const unsigned char* __restrict__ mask, // (B,C) jnp.bool_ = 1 byte
                                        const float* __restrict__ we_w, // (H)
                                        const float* __restrict__ we_b, // (1)
                                        float* __restrict__ attn) {     // (B*Q,C)
  const int bq = blockIdx.x;
  const int b  = bq / Q_;
  const int tid = threadIdx.x;

  __shared__ float mq_s[H_];
  __shared__ float we_s[H_];
  __shared__ float r_s[256];

  if (tid < H_) {
    mq_s[tid] = mq[(long)bq * H_ + tid];
    we_s[tid] = we_w[tid];
  }
  __syncthreads();

  const float web = we_b[0];
  const float* mcb = mc + (long)b * C_ * H_;
  const unsigned char* mrow = mask + (long)b * C_;

  float ev[4];
#pragma unroll
  for (int t = 0; t < 4; ++t) {
    const int c = tid + t * 256;
    const float4* row = (const float4*)(mcb + (long)c * H_);
    float acc = 0.0f;
#pragma unroll 8
    for (int h4 = 0; h4 < H_ / 4; ++h4) {
      const float4 m = row[h4];
      const int h = h4 * 4;
      acc += we_s[h + 0] * fast_tanh(m.x + mq_s[h + 0]);
      acc += we_s[h + 1] * fast_tanh(m.y + mq_s[h + 1]);
      acc += we_s[h + 2] * fast_tanh(m.z + mq_s[h + 2]);
      acc += we_s[h + 3] * fast_tanh(m.w + mq_s[h + 3]);
    }
    ev[t] = mrow[c] ? (acc + web) : -INFINITY;
  }

  // block max
  float lmax = fmaxf(fmaxf(ev[0], ev[1]), fmaxf(ev[2], ev[3]));
  r_s[tid] = lmax;
  __syncthreads();
  for (int s = 128; s > 0; s >>= 1) {
    if (tid < s) r_s[tid] = fmaxf(r_s[tid], r_s[tid + s]);
    __syncthreads();
  }
  const float maxv = r_s[0];
  __syncthreads();

  const bool all_masked = isinf(maxv) && (maxv < 0.0f);
  float p[4];
  float lsum = 0.0f;
#pragma unroll
  for (int t = 0; t < 4; ++t) {
    p[t] = all_masked ? 0.0f : __expf(ev[t] - maxv);  // v_exp_f32 path
    lsum += p[t];
  }
  r_s[tid] = lsum;
  __syncthreads();
  for (int s = 128; s > 0; s >>= 1) {
    if (tid < s) r_s[tid] += r_s[tid + s];
    __syncthreads();
  }
  const float sum = r_s[0];
  const float inv = (sum > 0.0f) ? (1.0f / sum) : 0.0f;

#pragma unroll
  for (int t = 0; t < 4; ++t) {
    const int c = tid + t * 256;
    attn[(long)bq * C_ + c] = p[t] * inv;
  }
}

// ---------------------------------------------------------------------------
extern "C" void kernel_launch(void* const* d_in, const int* in_sizes, int n_in,
                              void* d_out, int out_size, void* d_ws, size_t ws_size,
                              hipStream_t stream) {
  (void)in_sizes; (void)n_in; (void)out_size; (void)ws_size;

  const float* query   = (const float*)d_in[0];   // (B,Q,QD)
  const float* context = (const float*)d_in[1];   // (B,C,CD)
  const unsigned char* mask = (const unsigned char*)d_in[2];  // (B,C) bool
  const float* wq_w = (const float*)d_in[3];      // (H,QD)
  const float* wq_b = (const float*)d_in[4];      // (H)
  const float* wc_w = (const float*)d_in[5];      // (H,CD)
  const float* wc_b = (const float*)d_in[6];      // (H)
  const float* we_w = (const float*)d_in[7];      // (1,H)
  const float* we_b = (const float*)d_in[8];      // (1)
  const float* lo_w = (const float*)d_in[9];      // (QD, QD+CD) = (512,1024)
  const float* lo_b = (const float*)d_in[10];     // (QD)

  float* out  = (float*)d_out;                    // (B,Q,QD) first
  float* attn = out + (long)B_ * Q_ * QD_;        // then (B,Q,C)

  // workspace: mc (4096x128) | mq (1024x128) | wc (1024x512)  ~ 4.5 MB
  float* mc = (float*)d_ws;
  float* mq = mc + (long)B_ * C_ * H_;
  float* wc = mq + (long)B_ * Q_ * H_;

  const dim3 blk(256);

  // 1) mc = context @ wc_w^T + wc_b : M=4096, N=128, K=512 -> 2048 tiles
  wmma_gemm_f32_kernel<true, false, true, false>
      <<<dim3(2048 / 8), blk, 0, stream>>>(context, nullptr, wc_w, wc_b, mc,
                                           B_ * C_, H_, CD_, H_ / 16, 2048, 0, 0, 0);

  // 2) mq = query @ wq_w^T + wq_b : M=1024, N=128, K=512 -> 512 tiles
  wmma_gemm_f32_kernel<true, false, true, false>
      <<<dim3(512 / 8), blk, 0, stream>>>(query, nullptr, wq_w, wq_b, mq,
                                          B_ * Q_, H_, QD_, H_ / 16, 512, 0, 0, 0);

  // 3) emission + mask + softmax -> attn (written straight into d_out slice)
  emission_softmax_kernel<<<dim3(B_ * Q_), blk, 0, stream>>>(mc, mq, mask, we_w,
                                                             we_b, attn);

  // 4) wc[b] = attn[b] @ context[b] : M=256, N=512, K=1024, batched over B
  wmma_gemm_f32_kernel<false, false, false, false>
      <<<dim3(512 / 8, B_), blk, 0, stream>>>(attn, nullptr, context, nullptr, wc,
                                              Q_, CD_, C_, CD_ / 16, 512,
                                              (long)Q_ * C_, (long)C_ * CD_,
                                              (long)Q_ * CD_);

  // 5) out = tanh([wc | query] @ lo_w^T + lo_b) : M=1024, N=512, K=1024
  wmma_gemm_f32_kernel<true, true, true, true>
      <<<dim3(2048 / 8), blk, 0, stream>>>(wc, query, lo_w, lo_b, out,
                                           B_ * Q_, QD_, QD_ + CD_, QD_ / 16,
                                           2048, 0, 0, 0);
}